// KGAT_70806830842640
// MI455X (gfx1250) — compile-verified
//
#include <hip/hip_runtime.h>
#include <hip/hip_bf16.h>

// f32 ext-vector types matching CDNA5 WMMA fragment sizes for 16x16x4 f32:
// A (16x4) = 64 f32 / 32 lanes = 2 VGPRs; C/D (16x16) = 256 f32 / 32 lanes = 8 VGPRs.
typedef float f32x2 __attribute__((ext_vector_type(2)));
typedef float f32x8 __attribute__((ext_vector_type(8)));

#define NEG_SLOPE 0.01f
#define NORM_EPS  1e-12f

__device__ __forceinline__ void atomic_add_f32(float* p, float v) {
    // Lowers to GLOBAL_ATOMIC_ADD_F32 (resolved at the L2 atomic units).
    __hip_atomic_fetch_add(p, v, __ATOMIC_RELAXED, __HIP_MEMORY_SCOPE_AGENT);
}

// out[:, 0:64] = embeddings (first block of the concat)
__global__ void copy_emb_kernel(const float* __restrict__ emb,
                                float* __restrict__ out, int n) {
    int t = blockIdx.x * blockDim.x + threadIdx.x;
    if (t >= n * 64) return;
    int r = t >> 6;
    int j = t & 63;
    out[(size_t)r * 176 + j] = emb[t];
}

// side = A_coo @ ego : one thread per edge, float4 gathers from the
// L2-resident embedding table, f32 atomic scatter-add to the dest row.
template <int D>
__global__ void spmm_scatter_kernel(const float* __restrict__ vals,
                                    const int* __restrict__ rows,
                                    const int* __restrict__ cols,
                                    const float* __restrict__ ego,
                                    float* __restrict__ side,
                                    int nnz) {
    int e = blockIdx.x * blockDim.x + threadIdx.x;
    if (e >= nnz) return;
    // Stream-ahead prefetch on the edge arrays (global_prefetch_b8; speculative,
    // OOB translation failures silently dropped per ISA 7.3).
    __builtin_prefetch(vals + e + 8192, 0, 0);
    __builtin_prefetch(cols + e + 8192, 0, 0);

    float v = vals[e];
    int   r = rows[e];
    int   c = cols[e];
    const float4* __restrict__ src = (const float4*)(ego + (size_t)c * D);
    float* __restrict__ dst = side + (size_t)r * D;
#pragma unroll
    for (int j = 0; j < D / 4; ++j) {
        float4 m = src[j];
        atomic_add_f32(dst + 4 * j + 0, v * m.x);
        atomic_add_f32(dst + 4 * j + 1, v * m.y);
        atomic_add_f32(dst + 4 * j + 2, v * m.z);
        atomic_add_f32(dst + 4 * j + 3, v * m.w);
    }
}

// Bi-interaction layer via V_WMMA_F32_16X16X4_F32:
//   out = leaky((ego+side) @ W1 + b1) + leaky((ego*side) @ W2 + b2)
// One wave owns one N-tile (16 output cols) and caches the FULL W1/W2
// fragment strip for that tile in VGPRs (<= 64 VGPRs), then sweeps MT
// M-tiles reusing them. A-operands are loaded per k-step: the compiler
// builds a distance-1 software pipeline (loads for step k+1 issued behind
// the WMMAs of step k), which is the deepest pipelining that fits the
// VGPR budget WITHOUT spilling the B-cache to scratch (staging all A
// fragments was tried and forced scratch reloads of the weight cache).
// Fragment layouts per ISA 7.12.2 (32-bit A 16x4 / row-striped B / C-D 16x16).
template <int DIN, int DOUT, int MT>
__global__ void bi_gemm_wmma_kernel(const float* __restrict__ ego,
                                    const float* __restrict__ side,
                                    const float* __restrict__ W1,
                                    const float* __restrict__ B1,
                                    const float* __restrict__ W2,
                                    const float* __restrict__ B2,
                                    float* __restrict__ out,
                                    int n_rows) {
    constexpr int KSTEPS  = DIN / 4;
    constexpr int TILES_N = DOUT / 16;

    const int wave = (blockIdx.x * blockDim.x + threadIdx.x) >> 5;
    const int lane = threadIdx.x & 31;
    const int tiles_m = n_rows >> 4;
    const int mgroups = (tiles_m + MT - 1) / MT;
    if (wave >= mgroups * TILES_N) return;  // wave-uniform: EXEC all-1s for WMMA

    const int tn   = wave % TILES_N;
    const int tm0  = (wave / TILES_N) * MT;
    const int ncol  = tn * 16 + (lane & 15);
    const int khalf = (lane >> 4) << 1;     // 0 for lanes 0-15, 2 for lanes 16-31

    // Register-cache all B fragments for this column strip.
    f32x2 b1f[KSTEPS], b2f[KSTEPS];
#pragma unroll
    for (int ks = 0; ks < KSTEPS; ++ks) {
        const int kb = ks * 4 + khalf;
        b1f[ks] = f32x2{ W1[kb * DOUT + ncol], W1[(kb + 1) * DOUT + ncol] };
        b2f[ks] = f32x2{ W2[kb * DOUT + ncol], W2[(kb + 1) * DOUT + ncol] };
    }
    const float bias1 = B1[ncol];
    const float bias2 = B2[ncol];

    const int tm_end = min(tm0 + MT, tiles_m);
    for (int tm = tm0; tm < tm_end; ++tm) {
        const int mrow = tm * 16 + (lane & 15);
        const float* __restrict__ egoRow  = ego  + (size_t)mrow * DIN;
        const float* __restrict__ sideRow = side + (size_t)mrow * DIN;

        f32x8 acc1 = {};
        f32x8 acc2 = {};
#pragma unroll
        for (int ks = 0; ks < KSTEPS; ++ks) {
            const int kb = ks * 4 + khalf;          // even -> 8B-aligned f32x2
            f32x2 e = *(const f32x2*)(egoRow + kb);
            f32x2 s = *(const f32x2*)(sideRow + kb);
            f32x2 a1 = { e.x + s.x, e.y + s.y };    // sum-embedding operand
            f32x2 a2 = { e.x * s.x, e.y * s.y };    // bi-embedding operand
            acc1 = __builtin_amdgcn_wmma_f32_16x16x4_f32(false, a1, false, b1f[ks],
                                                         (short)0, acc1, false, false);
            acc2 = __builtin_amdgcn_wmma_f32_16x16x4_f32(false, a2, false, b2f[ks],
                                                         (short)0, acc2, false, false);
        }

        const int mbase = tm * 16 + ((lane >> 4) << 3);  // rows j / j+8 split
#pragma unroll
        for (int j = 0; j < 8; ++j) {
            float x1 = acc1[j] + bias1;
            float x2 = acc2[j] + bias2;
            x1 = (x1 >= 0.f) ? x1 : NEG_SLOPE * x1;
            x2 = (x2 >= 0.f) ? x2 : NEG_SLOPE * x2;
            // lanes 0-15 write 16 consecutive cols of row mbase+j -> coalesced
            out[(size_t)(mbase + j) * DOUT + ncol] = x1 + x2;
        }
    }
}

// Row-wise L2 normalize into the (N,176) output at a column offset.
// One wave32 per row, shuffle-tree reduction (warpSize == 32 on CDNA5).
template <int D>
__global__ void l2norm_out_kernel(const float* __restrict__ x,
                                  float* __restrict__ out,
                                  int n, int col_off) {
    int wave = (blockIdx.x * blockDim.x + threadIdx.x) >> 5;
    int lane = threadIdx.x & 31;
    if (wave >= n) return;
    const float* __restrict__ rowp = x + (size_t)wave * D;
    float s = 0.f;
#pragma unroll
    for (int j = lane; j < D; j += 32) {
        float v = rowp[j];
        s += v * v;
    }
#pragma unroll
    for (int off = 16; off > 0; off >>= 1)
        s += __shfl_xor(s, off, 32);
    float inv = 1.0f / fmaxf(sqrtf(s), NORM_EPS);
    float* __restrict__ orow = out + (size_t)wave * 176 + col_off;
#pragma unroll
    for (int j = lane; j < D; j += 32)
        orow[j] = rowp[j] * inv;
}

extern "C" void kernel_launch(void* const* d_in, const int* in_sizes, int n_in,
                              void* d_out, int out_size, void* d_ws, size_t ws_size,
                              hipStream_t stream) {
    const float* emb   = (const float*)d_in[0];
    const float* avals = (const float*)d_in[1];
    const float* w1_0 = (const float*)d_in[2];
    const float* b1_0 = (const float*)d_in[3];
    const float* w2_0 = (const float*)d_in[4];
    const float* b2_0 = (const float*)d_in[5];
    const float* w1_1 = (const float*)d_in[6];
    const float* b1_1 = (const float*)d_in[7];
    const float* w2_1 = (const float*)d_in[8];
    const float* b2_1 = (const float*)d_in[9];
    const float* w1_2 = (const float*)d_in[10];
    const float* b1_2 = (const float*)d_in[11];
    const float* w2_2 = (const float*)d_in[12];
    const float* b2_2 = (const float*)d_in[13];
    const int* arows = (const int*)d_in[14];
    const int* acols = (const int*)d_in[15];

    const int N   = in_sizes[0] / 64;   // 170000 (divisible by 16)
    const int nnz = in_sizes[1];
    float* out = (float*)d_out;

    // Workspace layout: ego1(64) | side(64) | ego2(32) | ego3(16) = 176*N floats
    float* ego1  = (float*)d_ws;
    float* sideb = ego1 + (size_t)N * 64;
    float* ego2  = sideb + (size_t)N * 64;
    float* ego3  = ego2 + (size_t)N * 32;

    constexpr int MT = 8;               // M-tiles swept per wave in the GEMM
    const int tiles_m = N / 16;
    const int mgroups = (tiles_m + MT - 1) / MT;

    // out[:, 0:64] = raw embeddings
    {
        int total = N * 64;
        copy_emb_kernel<<<(total + 255) / 256, 256, 0, stream>>>(emb, out, N);
    }

    // ---- Layer 0: 64 -> 64 ----
    hipMemsetAsync(sideb, 0, (size_t)N * 64 * sizeof(float), stream);
    spmm_scatter_kernel<64><<<(nnz + 255) / 256, 256, 0, stream>>>(
        avals, arows, acols, emb, sideb, nnz);
    {
        int waves = mgroups * (64 / 16);
        bi_gemm_wmma_kernel<64, 64, MT><<<(waves + 7) / 8, 256, 0, stream>>>(
            emb, sideb, w1_0, b1_0, w2_0, b2_0, ego1, N);
    }
    l2norm_out_kernel<64><<<(N + 7) / 8, 256, 0, stream>>>(ego1, out, N, 64);

    // ---- Layer 1: 64 -> 32 ----
    hipMemsetAsync(sideb, 0, (size_t)N * 64 * sizeof(float), stream);
    spmm_scatter_kernel<64><<<(nnz + 255) / 256, 256, 0, stream>>>(
        avals, arows, acols, ego1, sideb, nnz);
    {
        int waves = mgroups * (32 / 16);
        bi_gemm_wmma_kernel<64, 32, MT><<<(waves + 7) / 8, 256, 0, stream>>>(
            ego1, sideb, w1_1, b1_1, w2_1, b2_1, ego2, N);
    }
    l2norm_out_kernel<32><<<(N + 7) / 8, 256, 0, stream>>>(ego2, out, N, 128);

    // ---- Layer 2: 32 -> 16 ----
    hipMemsetAsync(sideb, 0, (size_t)N * 32 * sizeof(float), stream);
    spmm_scatter_kernel<32><<<(nnz + 255) / 256, 256, 0, stream>>>(
        avals, arows, acols, ego2, sideb, nnz);
    {
        int waves = mgroups * (16 / 16);
        bi_gemm_wmma_kernel<32, 16, MT><<<(waves + 7) / 8, 256, 0, stream>>>(
            ego2, sideb, w1_2, b1_2, w2_2, b2_2, ego3, N);
    }
    l2norm_out_kernel<16><<<(N + 7) / 8, 256, 0, stream>>>(ego3, out, N, 160);
}